// DistortionReg_90348932038893
// MI455X (gfx1250) — compile-verified
//
#include <hip/hip_runtime.h>

// DistortionReg on MI455X (gfx1250, wave32).
// loss_inter as WS[N,32] x DUS[32,32] GEMM via V_WMMA_F32_16X16X4_F32,
// DUS synthesized in registers from a constexpr bin-midpoint table.
// Memory-bound: one full read of ws (265 MB) ~= 11.4 us floor at 23.3 TB/s.

#define NRAYS   (1920 * 1080)
#define BINS    32
#define TILES   (NRAYS / 16)      // 129600 tiles of 16 rays
#define BLOCKS  1296
#define TPB     256
#define WPB     (TPB / 32)        // 8 waves per block

typedef __attribute__((ext_vector_type(2))) float v2f;
typedef __attribute__((ext_vector_type(8))) float v8f;

#define NEARV 0.2f
#define FARV  1000.0f
#define EPSV  1.1920928955078125e-07f   // FLT_EPSILON

struct Tab { float us[BINS]; float ds[BINS]; };

constexpr Tab make_tab() {
    Tab tab{};
    float s[BINS + 1] = {};
    const float t0 = NEARV + EPSV;
    const float g0 = 1.0f / t0;
    const float gf = 1.0f / FARV;
    const float denom = gf - g0;
    const float step = (FARV - t0) / (float)BINS;   // jnp.linspace step
    for (int i = 0; i <= BINS; ++i) {
        const float ti = t0 + step * (float)i;
        s[i] = (1.0f / ti - g0) / denom;
    }
    for (int i = 0; i < BINS; ++i) {
        tab.us[i] = 0.5f * (s[i + 1] + s[i]);
        tab.ds[i] = s[i + 1] - s[i];
    }
    return tab;
}

__constant__ Tab c_tab = make_tab();

// Phase 1: per-wave 16-ray tiles. T = WS_tile(16x32) @ DUS(32x32) via
// 8 chained K=4 f32 WMMAs per 16-col tile (2 tiles), then per-ray dot
// with WS and a full-wave butterfly reduction. One fp32 partial per block.
__global__ __launch_bounds__(TPB)
void distortion_phase1(const float* __restrict__ ws, float* __restrict__ partial) {
    const int lane  = threadIdx.x & 31;
    const int wave  = threadIdx.x >> 5;
    const int gwave = blockIdx.x * WPB + wave;
    const int nwave = gridDim.x * WPB;
    const int h = lane >> 4;      // lane half: selects K pair {2h, 2h+1}
    const int j = lane & 15;      // row (A) / column (B,C,D) index

    const float u_j0 = c_tab.us[j];
    const float u_j1 = c_tab.us[16 + j];

    float inter_acc = 0.0f;
    float intra_acc = 0.0f;

    for (int tile = gwave; tile < TILES; tile += nwave) {
        const int r0 = tile * 16;
        const float* arow = ws + (size_t)(r0 + j) * BINS;  // this lane's A row

        v8f acc0 = {};   // T columns  0..15
        v8f acc1 = {};   // T columns 16..31

#pragma unroll
        for (int k = 0; k < 8; ++k) {
            const int c = 4 * k + 2 * h;                   // K columns {c, c+1}
            const v2f a = *(const v2f*)(arow + c);         // 8B aligned

            // intra term: each ws element is touched exactly once per wave
            intra_acc += a.x * a.x * c_tab.ds[c] + a.y * a.y * c_tab.ds[c + 1];

            // B fragment of DUS, built in registers (mirrors the A K-layout)
            const float uc0 = c_tab.us[c];
            const float uc1 = c_tab.us[c + 1];
            v2f b0, b1;
            b0.x = __builtin_fabsf(uc0 - u_j0);
            b0.y = __builtin_fabsf(uc1 - u_j0);
            b1.x = __builtin_fabsf(uc0 - u_j1);
            b1.y = __builtin_fabsf(uc1 - u_j1);

            acc0 = __builtin_amdgcn_wmma_f32_16x16x4_f32(
                       false, a, false, b0, (short)0, acc0, false, false);
            acc1 = __builtin_amdgcn_wmma_f32_16x16x4_f32(
                       false, a, false, b1, (short)0, acc1, false, false);
        }

        // C/D layout: lane (h,j), VGPR r holds T[row r + 8h][col j (+16)].
        // inter[m] = sum_n ws[m,n] * T[m,n]; rows hot in WGP$/L2 from A loads.
        float psum = 0.0f;
#pragma unroll
        for (int r = 0; r < 8; ++r) {
            const size_t m = (size_t)(r0 + r + 8 * h) * BINS;
            psum += ws[m + j]      * acc0[r];
            psum += ws[m + 16 + j] * acc1[r];
        }
        inter_acc += psum;   // summing lanes later covers all 16 rays x 32 cols
    }

    float total = inter_acc + intra_acc * (1.0f / 3.0f);
#pragma unroll
    for (int off = 16; off >= 1; off >>= 1)
        total += __shfl_xor(total, off, 32);

    __shared__ float s_part[WPB];
    if (lane == 0) s_part[wave] = total;
    __syncthreads();
    if (threadIdx.x == 0) {
        float b = 0.0f;
#pragma unroll
        for (int w = 0; w < WPB; ++w) b += s_part[w];
        partial[blockIdx.x] = b;
    }
}

// Phase 2: deterministic final reduction -> mean.
__global__ __launch_bounds__(TPB)
void distortion_phase2(const float* __restrict__ partial, float* __restrict__ out, int n) {
    float s = 0.0f;
    for (int i = threadIdx.x; i < n; i += TPB) s += partial[i];
#pragma unroll
    for (int off = 16; off >= 1; off >>= 1)
        s += __shfl_xor(s, off, 32);

    __shared__ float sp[WPB];
    if ((threadIdx.x & 31) == 0) sp[threadIdx.x >> 5] = s;
    __syncthreads();
    if (threadIdx.x == 0) {
        float t = 0.0f;
#pragma unroll
        for (int w = 0; w < WPB; ++w) t += sp[w];
        out[0] = t * (1.0f / (float)NRAYS);
    }
}

extern "C" void kernel_launch(void* const* d_in, const int* in_sizes, int n_in,
                              void* d_out, int out_size, void* d_ws, size_t ws_size,
                              hipStream_t stream) {
    (void)in_sizes; (void)n_in; (void)out_size; (void)ws_size;
    const float* ws  = (const float*)d_in[0];
    float* partial   = (float*)d_ws;          // BLOCKS floats of scratch
    float* out       = (float*)d_out;

    distortion_phase1<<<BLOCKS, TPB, 0, stream>>>(ws, partial);
    distortion_phase2<<<1, TPB, 0, stream>>>(partial, out, BLOCKS);
}